// TRESCAL_84524956385797
// MI455X (gfx1250) — compile-verified
//
#include <hip/hip_runtime.h>

typedef __attribute__((ext_vector_type(2))) float v2f;
typedef __attribute__((ext_vector_type(8))) float v8f;

#define RANK 128
#define NENT 40000
#define NB   32     // batch size == wave32 width
#define EPB  64     // entities per block in main kernel

// ---------------------------------------------------------------------------
// Kernel A: gather t[b,:] = te_emb[time[b]] and compute
//           lhs[b,:] = ent[sub] * sin(W[sub]*t + b[sub])
// grid: (NB) blocks x RANK threads
// ---------------------------------------------------------------------------
__global__ void tk_prep(const int* __restrict__ x,
                        const float* __restrict__ ent,
                        const float* __restrict__ te,
                        const float* __restrict__ W,
                        const float* __restrict__ bvec,
                        float* __restrict__ t_ws,
                        float* __restrict__ lhs_ws) {
    int b = blockIdx.x;
    int j = threadIdx.x;
    int sub  = x[b * 4 + 0];
    int time = x[b * 4 + 3];
    float tv = te[(size_t)time * RANK + j];
    size_t ro = (size_t)sub * RANK + j;
    float lv = ent[ro] * __sinf(fmaf(W[ro], tv, bvec[ro]));
    t_ws[b * RANK + j]   = tv;
    lhs_ws[b * RANK + j] = lv;
}

// ---------------------------------------------------------------------------
// Kernel B: q[b,:] = lhs[b,:] @ rel_emb[rel_i[b]]  (1x128 * 128x128)
// via V_WMMA_F32_16X16X4_F32. One wave per (batch, 16-wide N tile):
// grid NB blocks x 256 threads (8 waves -> 8 N tiles), K advanced 4 per WMMA.
// A-matrix: only M=0 row holds lhs. Masking is done ARITHMETICALLY
// (multiply by 0/1 per lane) so the compiler emits unconditional clause
// loads instead of exec-mask-predicated branches around the WMMA.
// ---------------------------------------------------------------------------
__global__ void tk_q(const int* __restrict__ x,
                     const float* __restrict__ rel_emb,
                     const float* __restrict__ lhs_ws,
                     float* __restrict__ q_ws) {
    int b    = blockIdx.x;
    int wave = threadIdx.x >> 5;
    int lane = threadIdx.x & 31;
    int n0   = wave * 16;

    const float* rel = rel_emb + (size_t)x[b * 4 + 1] * (RANK * RANK);
    const float* lhs = lhs_ws + b * RANK;

    int hf = lane >> 4;           // 0: K pair {kb,kb+1}, 1: K pair {kb+2,kb+3}
    int nn = n0 + (lane & 15);    // B-matrix column this lane provides
    // 1.0f on lanes carrying A-matrix row M=0 (lanes 0 and 16), else 0.0f.
    float m0f = ((lane & 15) == 0) ? 1.0f : 0.0f;

    v8f c = {};
    for (int kb = 0; kb < RANK; kb += 4) {
        int krow = kb + 2 * hf;
        // Unconditional loads in every lane (always in-bounds), masked by mul.
        float ax = lhs[krow];
        float ay = lhs[krow + 1];
        v2f a;
        a.x = ax * m0f;
        a.y = ay * m0f;
        v2f bm;
        bm.x = rel[(size_t)krow * RANK + nn];
        bm.y = rel[(size_t)(krow + 1) * RANK + nn];
        c = __builtin_amdgcn_wmma_f32_16x16x4_f32(
                /*neg_a=*/false, a, /*neg_b=*/false, bm,
                /*c_mod=*/(short)0, c, /*reuse_a=*/false, /*reuse_b=*/false);
    }
    // D layout: VGPR0, lanes 0..15 hold (M=0, N=lane) -> q[b][n0+lane]
    if (lane < 16) q_ws[b * RANK + n0 + lane] = c[0];
}

// ---------------------------------------------------------------------------
// Kernel C (the heavy one): pred[b,n] = sum_j q[b,j]*ent[n,j]*sin(W[n,j]*t[b,j]+b[n,j])
// lanes = batches (wave32 == B), so no cross-lane reduction.
// t,q staged in LDS transposed [j][b] with stride 33 (bank-conflict-free).
// Entity rows are wave-uniform float4 streams. 625 blocks x 8 waves x 8 ents.
// ---------------------------------------------------------------------------
__global__ __launch_bounds__(256) void tk_main(const float* __restrict__ ent,
                                               const float* __restrict__ W,
                                               const float* __restrict__ bvec,
                                               const float* __restrict__ t_ws,
                                               const float* __restrict__ q_ws,
                                               float* __restrict__ out) {
    __shared__ float t_s[RANK * 33];
    __shared__ float q_s[RANK * 33];

    int tid = threadIdx.x;
    for (int i = tid; i < NB * RANK; i += 256) {
        int bb = i >> 7;          // batch
        int j  = i & (RANK - 1);  // feature
        t_s[j * 33 + bb] = t_ws[i];
        q_s[j * 33 + bb] = q_ws[i];
    }
    __syncthreads();

    int wave = tid >> 5;
    int lane = tid & 31;          // lane == batch index b
    int nbase = blockIdx.x * EPB + wave * 8;

    for (int i = 0; i < 8; ++i) {
        int n = nbase + i;
        const float* er = ent  + (size_t)n * RANK;
        const float* wr = W    + (size_t)n * RANK;
        const float* br = bvec + (size_t)n * RANK;
        // hint the next entity's rows into cache while we grind sins
        __builtin_prefetch(er + RANK, 0, 1);
        __builtin_prefetch(wr + RANK, 0, 1);
        __builtin_prefetch(br + RANK, 0, 1);

        float acc = 0.0f;
#pragma unroll 4
        for (int j4 = 0; j4 < RANK; j4 += 4) {
            float4 e4 = *(const float4*)(er + j4);
            float4 w4 = *(const float4*)(wr + j4);
            float4 b4 = *(const float4*)(br + j4);
            {
                float tv = t_s[(j4 + 0) * 33 + lane];
                float qv = q_s[(j4 + 0) * 33 + lane];
                acc = fmaf(qv * e4.x, __sinf(fmaf(w4.x, tv, b4.x)), acc);
            }
            {
                float tv = t_s[(j4 + 1) * 33 + lane];
                float qv = q_s[(j4 + 1) * 33 + lane];
                acc = fmaf(qv * e4.y, __sinf(fmaf(w4.y, tv, b4.y)), acc);
            }
            {
                float tv = t_s[(j4 + 2) * 33 + lane];
                float qv = q_s[(j4 + 2) * 33 + lane];
                acc = fmaf(qv * e4.z, __sinf(fmaf(w4.z, tv, b4.z)), acc);
            }
            {
                float tv = t_s[(j4 + 3) * 33 + lane];
                float qv = q_s[(j4 + 3) * 33 + lane];
                acc = fmaf(qv * e4.w, __sinf(fmaf(w4.w, tv, b4.w)), acc);
            }
        }
        out[(size_t)lane * NENT + n] = acc;  // pred[b][n]
    }
}

// ---------------------------------------------------------------------------
extern "C" void kernel_launch(void* const* d_in, const int* in_sizes, int n_in,
                              void* d_out, int out_size, void* d_ws, size_t ws_size,
                              hipStream_t stream) {
    const int*   x    = (const int*)d_in[0];
    const float* ent  = (const float*)d_in[1];
    const float* rel  = (const float*)d_in[2];
    const float* te   = (const float*)d_in[3];
    const float* W    = (const float*)d_in[4];
    const float* bvec = (const float*)d_in[5];
    float* out = (float*)d_out;

    float* t_ws   = (float*)d_ws;            // 32*128 floats
    float* lhs_ws = t_ws + NB * RANK;        // 32*128 floats
    float* q_ws   = lhs_ws + NB * RANK;      // 32*128 floats

    tk_prep<<<NB, RANK, 0, stream>>>(x, ent, te, W, bvec, t_ws, lhs_ws);
    tk_q<<<NB, 256, 0, stream>>>(x, rel, lhs_ws, q_ws);
    tk_main<<<NENT / EPB, 256, 0, stream>>>(ent, W, bvec, t_ws, q_ws, out);
}